// PointPillarScatter3d_2860448219531
// MI455X (gfx1250) — compile-verified
//
#include <hip/hip_runtime.h>
#include <stddef.h>

typedef __attribute__((ext_vector_type(4))) float f4;
typedef __attribute__((ext_vector_type(4))) int   i4;

#define PPS_NX  512
#define PPS_NY  512
#define PPS_NZ  2
#define PPS_C   64                              // features per pillar
#define PPS_P   (PPS_NZ * PPS_NY * PPS_NX)      // 524288 cells per batch
#define PPS_SPB (PPS_P / 256)                   // 2048 s-blocks per batch image

// ---------------- Pass 1: map[cell] = -1 (8 MB for B=4; L2 resident) ------
__global__ void pps_fill_map(i4* __restrict__ map4, int n4) {
    int i = blockIdx.x * blockDim.x + threadIdx.x;
    if (i < n4) {
        i4 m1 = { -1, -1, -1, -1 };
        map4[i] = m1;
    }
}

// ---------------- Pass 2: map[b*P + z*NY*NX + y*NX + x] = pillar index ----
__global__ void pps_scatter_idx(const i4* __restrict__ coords4,
                                int* __restrict__ map, int N) {
    int n = blockIdx.x * blockDim.x + threadIdx.x;
    if (n >= N) return;
    i4 c = coords4[n];  // (b, z, y, x)
    int cell = c.x * PPS_P + c.y * (PPS_NY * PPS_NX) + c.z * PPS_NX + c.w;
    map[cell] = n;
}

// ---------------- Pass 3: stream the dense output, fully coalesced --------
// Block = 256 consecutive cells s of one batch b. Loop over 64 channels:
// each store instruction writes 128B contiguous per wave (lane = s).
// Feature rows (38.4 MB total) gathered as float4 -> L2 resident.
// Output written with non-temporal hint: 512 MB stream, never re-read.
__global__ void pps_expand(const f4* __restrict__ feat4,
                           const int* __restrict__ map,
                           float* __restrict__ out) {
    int b = blockIdx.x / PPS_SPB;
    int s = (blockIdx.x % PPS_SPB) * 256 + threadIdx.x;

    int idx = map[(size_t)b * PPS_P + s];
    bool valid = idx >= 0;
    // invalid lanes read row 0 (always cache-hit) and select 0 afterwards:
    // unconditional loads, no divergence on the load path.
    const f4* frow = feat4 + (size_t)(valid ? idx : 0) * (PPS_C / 4);
    __builtin_prefetch((const void*)frow, 0, 1);                       // global_prefetch_b8
    __builtin_prefetch((const void*)((const char*)frow + 128), 0, 1);  // row = 256B

    float* outp = out + (size_t)b * ((size_t)PPS_C * PPS_P) + (size_t)s;

#pragma unroll
    for (int c4 = 0; c4 < PPS_C / 4; ++c4) {
        f4 t = frow[c4];
        float v0 = valid ? t.x : 0.0f;
        float v1 = valid ? t.y : 0.0f;
        float v2 = valid ? t.z : 0.0f;
        float v3 = valid ? t.w : 0.0f;
        __builtin_nontemporal_store(v0, outp + (size_t)(4 * c4 + 0) * PPS_P);
        __builtin_nontemporal_store(v1, outp + (size_t)(4 * c4 + 1) * PPS_P);
        __builtin_nontemporal_store(v2, outp + (size_t)(4 * c4 + 2) * PPS_P);
        __builtin_nontemporal_store(v3, outp + (size_t)(4 * c4 + 3) * PPS_P);
    }
}

// ---------------- Fallback path (workspace too small) ---------------------
__global__ void pps_zero(f4* __restrict__ out4, long long n4) {
    long long i = (long long)blockIdx.x * blockDim.x + threadIdx.x;
    long long stride = (long long)gridDim.x * blockDim.x;
    f4 z = { 0.0f, 0.0f, 0.0f, 0.0f };
    for (; i < n4; i += stride)
        __builtin_nontemporal_store(z, out4 + i);
}

__global__ void pps_scatter_feat(const f4* __restrict__ feat4,
                                 const i4* __restrict__ coords4,
                                 float* __restrict__ out, int N) {
    int n = blockIdx.x * blockDim.x + threadIdx.x;
    if (n >= N) return;
    i4 c = coords4[n];
    size_t base = (size_t)c.x * ((size_t)PPS_C * PPS_P)
                + (size_t)(c.y * (PPS_NY * PPS_NX) + c.z * PPS_NX + c.w);
    const f4* frow = feat4 + (size_t)n * (PPS_C / 4);
#pragma unroll
    for (int c4 = 0; c4 < PPS_C / 4; ++c4) {
        f4 t = frow[c4];
        out[base + (size_t)(4 * c4 + 0) * PPS_P] = t.x;
        out[base + (size_t)(4 * c4 + 1) * PPS_P] = t.y;
        out[base + (size_t)(4 * c4 + 2) * PPS_P] = t.z;
        out[base + (size_t)(4 * c4 + 3) * PPS_P] = t.w;
    }
}

extern "C" void kernel_launch(void* const* d_in, const int* in_sizes, int n_in,
                              void* d_out, int out_size, void* d_ws, size_t ws_size,
                              hipStream_t stream) {
    (void)n_in;
    const float* feat   = (const float*)d_in[0];   // [N, 64] f32
    const int*   coords = (const int*)d_in[1];     // [N, 4]  i32 (b,z,y,x)
    float*       out    = (float*)d_out;           // [B, 128, 512, 512] f32

    const int N = in_sizes[1] / 4;
    int B = (int)((long long)out_size / ((long long)PPS_C * (long long)PPS_P));
    if (B < 1) B = 1;

    const size_t mapBytes = (size_t)B * (size_t)PPS_P * sizeof(int);
    if (d_ws != nullptr && ws_size >= mapBytes) {
        int* map = (int*)d_ws;
        const int n4 = (int)(((size_t)B * (size_t)PPS_P) / 4);
        pps_fill_map<<<(n4 + 255) / 256, 256, 0, stream>>>((i4*)map, n4);
        pps_scatter_idx<<<(N + 255) / 256, 256, 0, stream>>>(
            (const i4*)coords, map, N);
        pps_expand<<<B * PPS_SPB, 256, 0, stream>>>(
            (const f4*)feat, map, out);
    } else {
        const long long n4 = (long long)out_size / 4;
        pps_zero<<<4096, 256, 0, stream>>>((f4*)out, n4);
        pps_scatter_feat<<<(N + 255) / 256, 256, 0, stream>>>(
            (const f4*)feat, (const i4*)coords, out, N);
    }
}